// PostProcessing4_90263032692906
// MI455X (gfx1250) — compile-verified
//
#include <hip/hip_runtime.h>
#include <hip/hip_bf16.h>
#include <cmath>

// ---------------------------------------------------------------------------
// PostProcessing (panoptic post-proc) for MI455X / gfx1250, wave32.
// B=2, H=512, W=1024, C=34 classes, TOP_K=200, circle radius 5, pool 7x7.
//
// Design (bandwidth-first):
//  * logits (142.6 MB) read exactly once: fused softmax + argmax +
//    per-(instance,class) accumulation in LDS -> sparse global atomics.
//  * nearest-center argmin as rank-3 GEMM on V_WMMA_F32_16X16X4_F32:
//    score = [px,py,1,0] x [-2cx; -2cy; cx^2+cy^2; 0]
//    2 A-tiles (32 px) per wave x 13 preloaded B-tiles -> 26 WMMAs/wave,
//    branchless b64 B loads (no EXEC juggling around the WMMA chain).
//  * vote-map halo staging via GLOBAL_LOAD_ASYNC_TO_LDS_B32 (ASYNCcnt) when
//    the builtin exists (guarded with __has_builtin; clean fallback).
//    NOTE: this toolchain declares the builtin with generic `int*` params.
// ---------------------------------------------------------------------------

#define BATCH 2
#define HH 512
#define WW 1024
#define HW (HH*WW)
#define NCLS 34
#define TOPK 200
#define NSEG (TOPK+1)
#define NPAD 208          // TOPK padded to 13 WMMA N-tiles of 16
#define CRAD 5
#define PRAD 3            // 7x7 pool
#define CAP 65536         // candidate capacity per batch

typedef __attribute__((ext_vector_type(2))) float v2f;
typedef __attribute__((ext_vector_type(8))) float v8f;

#if __has_builtin(__builtin_amdgcn_global_load_async_to_lds_b32)
#define ASYNC_LDS 1
#endif

__device__ __forceinline__ void wait_async_zero() {
#if __has_builtin(__builtin_amdgcn_s_wait_asynccnt)
    __builtin_amdgcn_s_wait_asynccnt(0);
#else
    asm volatile("s_wait_asynccnt 0x0" ::: "memory");
#endif
}

// ------------------------------- init --------------------------------------
__global__ void init_kernel(float* counts_g, float* probsum_g, int* cand_cnt, int n_acc) {
    int i = blockIdx.x * blockDim.x + threadIdx.x;
    if (i < n_acc) { counts_g[i] = 0.f; probsum_g[i] = 0.f; }
    if (i < BATCH) cand_cnt[i] = 0;
}

// ------------------------------ vote map -----------------------------------
// vote[y][x] = -(1/81) * sum_{(dy,dx) in circle} sqrt((dx-rx)^2+(dy-ry)^2) - 1
// with zero-padded regression reads (matches jnp.pad).
__global__ void __launch_bounds__(256) vote_kernel(const float* __restrict__ reg,
                                                   float* __restrict__ vote) {
    const int TX = 32, TY = 8;
    __shared__ float s0[TY + 2*CRAD][TX + 2*CRAD];
    __shared__ float s1[TY + 2*CRAD][TX + 2*CRAD];
    int b  = blockIdx.z;
    int bx = blockIdx.x * TX, by = blockIdx.y * TY;
    const float* r0 = reg + (size_t)b * 2 * HW;
    const float* r1 = r0 + HW;
    int tid = threadIdx.y * TX + threadIdx.x;
    const int LW = TX + 2*CRAD, LH = TY + 2*CRAD;
    for (int i = tid; i < LW * LH; i += TX * TY) {
        int ly = i / LW, lx = i % LW;
        int gy = by + ly - CRAD, gx = bx + lx - CRAD;
        bool in = (gx >= 0 && gx < WW && gy >= 0 && gy < HH);
        int gp = gy * WW + gx;
#ifdef ASYNC_LDS
        if (in) {
            __builtin_amdgcn_global_load_async_to_lds_b32(
                (int*)(r0 + gp), (int*)&s0[ly][lx], 0, 0);
            __builtin_amdgcn_global_load_async_to_lds_b32(
                (int*)(r1 + gp), (int*)&s1[ly][lx], 0, 0);
        } else {
            s0[ly][lx] = 0.f;
            s1[ly][lx] = 0.f;
        }
#else
        s0[ly][lx] = in ? r0[gp] : 0.f;
        s1[ly][lx] = in ? r1[gp] : 0.f;
#endif
    }
#ifdef ASYNC_LDS
    wait_async_zero();
#endif
    __syncthreads();
    int lx = threadIdx.x + CRAD, ly = threadIdx.y + CRAD;
    float acc = 0.f;
    #pragma unroll
    for (int dy = -CRAD; dy <= CRAD; ++dy) {
        #pragma unroll
        for (int dx = -CRAD; dx <= CRAD; ++dx) {
            if (dx*dx + dy*dy > CRAD*CRAD) continue;   // compile-time circle mask
            float rx = s0[ly + dy][lx + dx];
            float ry = s1[ly + dy][lx + dx];
            float ex = (float)dx - rx, ey = (float)dy - ry;
            acc += sqrtf(ex*ex + ey*ey);
        }
    }
    int gp = (by + threadIdx.y) * WW + (bx + threadIdx.x);
    vote[(size_t)b * HW + gp] = -(acc / 81.0f) - 1.0f;
}

// ----------------------- 7x7 maxpool + candidate scan ----------------------
__global__ void __launch_bounds__(256) pool_kernel(const float* __restrict__ vote,
                                                   float* __restrict__ center_map,
                                                   float* __restrict__ cand_vals,
                                                   int* __restrict__ cand_idx,
                                                   int* __restrict__ cand_cnt,
                                                   float gate) {
    const int TX = 32, TY = 8;
    __shared__ float sv[TY + 2*PRAD][TX + 2*PRAD];
    int b  = blockIdx.z;
    int bx = blockIdx.x * TX, by = blockIdx.y * TY;
    const float* vb = vote + (size_t)b * HW;
    int tid = threadIdx.y * TX + threadIdx.x;
    const int LW = TX + 2*PRAD, LH = TY + 2*PRAD;
    for (int i = tid; i < LW * LH; i += TX * TY) {
        int ly = i / LW, lx = i % LW;
        int gy = by + ly - PRAD, gx = bx + lx - PRAD;
        bool in = (gx >= 0 && gx < WW && gy >= 0 && gy < HH);
        sv[ly][lx] = in ? vb[gy * WW + gx] : -INFINITY;   // SAME pad with -inf
    }
    __syncthreads();
    int lx = threadIdx.x + PRAD, ly = threadIdx.y + PRAD;
    float mx = -INFINITY;
    #pragma unroll
    for (int dy = -PRAD; dy <= PRAD; ++dy)
        #pragma unroll
        for (int dx = -PRAD; dx <= PRAD; ++dx)
            mx = fmaxf(mx, sv[ly + dy][lx + dx]);
    float v = sv[ly][lx];
    bool keep = (mx == v) && (v > gate);
    int gp = (by + threadIdx.y) * WW + (bx + threadIdx.x);
    center_map[(size_t)b * HW + gp] = keep ? v : 0.f;
    if (keep) {
        int pos = atomicAdd(&cand_cnt[b], 1);
        if (pos < CAP) {
            cand_vals[(size_t)b * CAP + pos] = v;
            cand_idx [(size_t)b * CAP + pos] = gp;
        }
    }
}

// -------------------------- top-200 selection ------------------------------
// One block per batch. Exact (value desc, index asc) ordering to match
// jax.lax.top_k stability; -inf slots filled with ascending unused indices.
__global__ void __launch_bounds__(256) topk_kernel(float* __restrict__ cand_vals,
                                                   const int* __restrict__ cand_idx,
                                                   const int* __restrict__ cand_cnt,
                                                   float* __restrict__ top_vals_out,
                                                   int* __restrict__ centers_out,
                                                   float* __restrict__ bm4) {
    __shared__ float sv[256];
    __shared__ int   si[256];
    __shared__ int   sp[256];
    __shared__ float s_cx[NPAD], s_cy[NPAD];
    __shared__ int   sel_idx[TOPK];
    __shared__ int   s_nvalid;
    int b = blockIdx.x, tid = threadIdx.x;
    int n = cand_cnt[b]; if (n > CAP) n = CAP;
    float* vals = cand_vals + (size_t)b * CAP;
    const int* idxs = cand_idx + (size_t)b * CAP;
    if (tid == 0) s_nvalid = 0;
    __syncthreads();
    for (int it = 0; it < TOPK; ++it) {
        float bv = -INFINITY; int bi = 0x7FFFFFFF; int bp = -1;
        for (int j = tid; j < n; j += 256) {
            float v = vals[j]; int id = idxs[j];
            if (v > bv || (v == bv && id < bi)) { bv = v; bi = id; bp = j; }
        }
        sv[tid] = bv; si[tid] = bi; sp[tid] = bp;
        __syncthreads();
        for (int s = 128; s > 0; s >>= 1) {
            if (tid < s) {
                if (sv[tid+s] > sv[tid] || (sv[tid+s] == sv[tid] && si[tid+s] < si[tid])) {
                    sv[tid] = sv[tid+s]; si[tid] = si[tid+s]; sp[tid] = sp[tid+s];
                }
            }
            __syncthreads();
        }
        if (tid == 0 && sp[0] >= 0) {
            float v = sv[0]; int id = si[0];
            top_vals_out[b * TOPK + it] = v;
            centers_out[(b * TOPK + it) * 2 + 0] = id / WW;
            centers_out[(b * TOPK + it) * 2 + 1] = id % WW;
            s_cx[it] = (float)(id % WW);
            s_cy[it] = (float)(id / WW);
            sel_idx[it] = id;
            vals[sp[0]] = -INFINITY;       // remove from pool
            s_nvalid = it + 1;
        }
        __syncthreads();
        if (sv[0] == -INFINITY) break;     // uniform across block
    }
    __syncthreads();
    if (tid == 0) {
        int nv = s_nvalid;
        int fill = 0;
        for (int slot = nv; slot < TOPK; ++slot) {
            for (;;) {
                bool used = false;
                for (int k = 0; k < nv; ++k) if (sel_idx[k] == fill) { used = true; break; }
                if (!used) break;
                ++fill;
            }
            top_vals_out[b * TOPK + slot] = -INFINITY;
            centers_out[(b * TOPK + slot) * 2 + 0] = fill / WW;
            centers_out[(b * TOPK + slot) * 2 + 1] = fill % WW;
            s_cx[slot] = 1.0e9f; s_cy[slot] = 1.0e9f;   // BIG sentinel
            ++fill;
        }
        for (int k = TOPK; k < NPAD; ++k) { s_cx[k] = 1.0e9f; s_cy[k] = 1.0e9f; }
    }
    __syncthreads();
    // B-matrix, 4 rows per center (K=3 row is zero), center-major so a lane
    // fetches its (k,k+1) pair with one aligned b64 load.
    if (tid < NPAD) {
        float cx = s_cx[tid], cy = s_cy[tid];
        float* d = bm4 + (size_t)b * NPAD * 4 + tid * 4;
        d[0] = -2.0f * cx;
        d[1] = -2.0f * cy;
        d[2] = cx * cx + cy * cy;
        d[3] = 0.0f;
    }
}

// --------------------- WMMA nearest-center argmin --------------------------
// Each wave: 32 pixels (two 16x4 A tiles).  B tiles (4x16) preloaded into 13
// v2f registers (branchless b64 loads) -> 26 back-to-back WMMAs per wave.
// Exact (val, idx) lexicographic min matches jnp.argmin tie-breaking.
__global__ void __launch_bounds__(256) nearest_center_kernel(
        const float* __restrict__ reg, const float* __restrict__ bm4,
        unsigned char* __restrict__ ccidx) {
    int b = blockIdx.y;
    int lane = threadIdx.x & 31;
    int wave = threadIdx.x >> 5;
    int pixelBase = (blockIdx.x * 8 + wave) * 32;
    int m  = lane & 15;       // A row / D column within half
    int hi = lane >> 4;       // which K/M half
    v2f a0, a1;
    if (hi == 0) {            // lanes 0-15: K=0,1 -> px, py
        int p0 = pixelBase + m;
        int p1 = pixelBase + 16 + m;
        const float* rx = reg + (size_t)b * 2 * HW;
        const float* ry = rx + HW;
        a0[0] = (float)((p0 & (WW - 1)) + 1) - rx[p0];
        a0[1] = (float)((p0 >> 10) + 1)      - ry[p0];
        a1[0] = (float)((p1 & (WW - 1)) + 1) - rx[p1];
        a1[1] = (float)((p1 >> 10) + 1)      - ry[p1];
    } else {                  // lanes 16-31: K=2,3 -> 1, 0
        a0[0] = 1.0f; a0[1] = 0.0f;
        a1[0] = 1.0f; a1[1] = 0.0f;
    }
    // preload all 13 B tiles, branchless: lane grabs rows (2*hi, 2*hi+1) of
    // its column with one 8B load.
    const float* bb = bm4 + (size_t)b * NPAD * 4;
    v2f bt[13];
    #pragma unroll
    for (int t = 0; t < 13; ++t) {
        int nn = t * 16 + m;
        bt[t] = *(const v2f*)(bb + nn * 4 + 2 * hi);
    }
    float best[16]; int bidx[16];          // [0..7]=A-tile0 rows, [8..15]=tile1
    #pragma unroll
    for (int v = 0; v < 16; ++v) { best[v] = 3.0e38f; bidx[v] = 0x7FFFFFFF; }
    #pragma unroll
    for (int t = 0; t < 13; ++t) {
        v8f c = {};
        v8f d0 = __builtin_amdgcn_wmma_f32_16x16x4_f32(
            false, a0, false, bt[t], (short)0, c, false, false);
        v8f d1 = __builtin_amdgcn_wmma_f32_16x16x4_f32(
            false, a1, false, bt[t], (short)0, c, false, false);
        int ci = t * 16 + m;  // center index of this lane's D column
        #pragma unroll
        for (int v = 0; v < 8; ++v) {
            float x0 = d0[v];
            if (x0 < best[v] || (x0 == best[v] && ci < bidx[v])) { best[v] = x0; bidx[v] = ci; }
            float x1 = d1[v];
            if (x1 < best[8+v] || (x1 == best[8+v] && ci < bidx[8+v])) { best[8+v] = x1; bidx[8+v] = ci; }
        }
    }
    // exact argmin over the 16 D-columns: butterfly within each 16-lane group
    #pragma unroll
    for (int mask = 1; mask < 16; mask <<= 1) {
        #pragma unroll
        for (int v = 0; v < 16; ++v) {
            float ov = __shfl_xor(best[v], mask, 16);
            int   oi = __shfl_xor(bidx[v], mask, 16);
            if (ov < best[v] || (ov == best[v] && oi < bidx[v])) {
                best[v] = ov; bidx[v] = oi;
            }
        }
    }
    #pragma unroll
    for (int v = 0; v < 8; ++v) {
        if (m == v) {   // lane v of each half writes pixels v + 8*hi (+16)
            int p0 = pixelBase + v + 8 * hi;
            int c0 = bidx[v];
            ccidx[(size_t)b * HW + p0] = (unsigned char)(c0 < 255 ? c0 : 255);
            int p1 = pixelBase + 16 + v + 8 * hi;
            int c1 = bidx[8 + v];
            ccidx[(size_t)b * HW + p1] = (unsigned char)(c1 < 255 ? c1 : 255);
        }
    }
}

// -------------- fused softmax / argmax / instance statistics ---------------
__global__ void __launch_bounds__(256) seg_stats_kernel(
        const float* __restrict__ logits, const unsigned char* __restrict__ ccidx,
        int* __restrict__ inst_map, float* __restrict__ seg_map,
        float* __restrict__ counts_g, float* __restrict__ probsum_g) {
    __shared__ float s_cnt[NSEG * NCLS];   // 27336 B
    __shared__ float s_ps [NSEG * NCLS];   // 27336 B
    int b = blockIdx.y;
    for (int i = threadIdx.x; i < NSEG * NCLS; i += blockDim.x) { s_cnt[i] = 0.f; s_ps[i] = 0.f; }
    __syncthreads();
    const float* lg = logits + (size_t)b * NCLS * HW;
    int stride = gridDim.x * blockDim.x;
    for (int p = blockIdx.x * blockDim.x + threadIdx.x; p < HW; p += stride) {
        __builtin_prefetch(lg + p + stride, 0, 0);   // global_prefetch_b8
        float l[NCLS];
        float mx = -INFINITY; int am = 0;
        #pragma unroll
        for (int c = 0; c < NCLS; ++c) {
            float v = lg[(size_t)c * HW + p];
            l[c] = v;
            if (v > mx) { mx = v; am = c; }   // strict > : lowest-index ties
        }
        float se = 0.f;
        #pragma unroll
        for (int c = 0; c < NCLS; ++c) { float e = expf(l[c] - mx); l[c] = e; se += e; }
        float inv = 1.0f / se;
        int cc = ccidx[(size_t)b * HW + p]; if (cc > TOPK - 1) cc = TOPK - 1;
        int inst = (am >= 24) ? (cc + 1) : 0;        // "things" classes 24..33
        inst_map[(size_t)b * HW + p] = inst;
        seg_map [(size_t)b * HW + p] = (float)am;
        atomicAdd(&s_cnt[inst * NCLS + am], 1.0f);
        #pragma unroll
        for (int c = 0; c < NCLS; ++c) atomicAdd(&s_ps[inst * NCLS + c], l[c] * inv);
    }
    __syncthreads();
    float* cg = counts_g  + (size_t)b * NSEG * NCLS;
    float* pg = probsum_g + (size_t)b * NSEG * NCLS;
    for (int i = threadIdx.x; i < NSEG * NCLS; i += blockDim.x) {
        if (s_cnt[i] != 0.f) atomicAdd(&cg[i], s_cnt[i]);
        if (s_ps [i] != 0.f) atomicAdd(&pg[i], s_ps[i]);
    }
}

// ------------------------------ finalize -----------------------------------
__global__ void finalize_kernel(const float* __restrict__ counts_g,
                                const float* __restrict__ probsum_g,
                                int* __restrict__ inst_class,
                                float* __restrict__ inst_seg_prob,
                                int* __restrict__ inst_size) {
    int b = blockIdx.x;
    for (int s = threadIdx.x + 1; s <= TOPK; s += blockDim.x) {
        const float* cn = counts_g  + ((size_t)b * NSEG + s) * NCLS;
        const float* ps = probsum_g + ((size_t)b * NSEG + s) * NCLS;
        float tot = 0.f, best = -1.f; int bc = 0;
        #pragma unroll
        for (int c = 0; c < NCLS; ++c) {
            float v = cn[c]; tot += v;
            if (v > best) { best = v; bc = c; }   // lowest-index ties
        }
        int o = b * TOPK + (s - 1);
        inst_class[o]    = bc;
        inst_size[o]     = (int)tot;
        inst_seg_prob[o] = ps[bc] / fmaxf(tot, 1.0f);
    }
}

// ------------------------------- launch ------------------------------------
extern "C" void kernel_launch(void* const* d_in, const int* in_sizes, int n_in,
                              void* d_out, int out_size, void* d_ws, size_t ws_size,
                              hipStream_t stream) {
    const float* logits = (const float*)d_in[0];
    const float* regs   = (const float*)d_in[2];
    float* out = (float*)d_out;

    // output layout (flat, return order)
    size_t off = 0;
    int*   inst_map   = (int*)  (out + off); off += (size_t)BATCH * HW;
    float* seg_map    =          out + off;  off += (size_t)BATCH * HW;
    int*   centers    = (int*)  (out + off); off += (size_t)BATCH * TOPK * 2;
    float* top_vals   =          out + off;  off += (size_t)BATCH * TOPK;
    int*   inst_class = (int*)  (out + off); off += (size_t)BATCH * TOPK;
    float* inst_prob  =          out + off;  off += (size_t)BATCH * TOPK;
    int*   inst_size  = (int*)  (out + off); off += (size_t)BATCH * TOPK;
    float* center_map =          out + off;

    // workspace layout
    float* ws = (float*)d_ws;
    size_t o = 0;
    float* vote      = ws + o;           o += (size_t)BATCH * HW;
    float* cand_vals = ws + o;           o += (size_t)BATCH * CAP;
    int*   cand_idx  = (int*)(ws + o);   o += (size_t)BATCH * CAP;
    float* bm4       = ws + o;           o += (size_t)BATCH * NPAD * 4;
    float* counts_g  = ws + o;           o += (size_t)BATCH * NSEG * NCLS;
    float* probsum_g = ws + o;           o += (size_t)BATCH * NSEG * NCLS;
    int*   cand_cnt  = (int*)(ws + o);   o += 16;
    unsigned char* ccidx = (unsigned char*)(ws + o);

    // circle threshold (host, matches _circle_constants float order)
    float thr_acc = 0.f;
    for (int i = 0; i < 11; ++i)
        for (int j = 0; j < 11; ++j) {
            int dx = j - 5, dy = i - 5;
            if (dx*dx + dy*dy <= 25) thr_acc += sqrtf((float)(dx*dx + dy*dy));
        }
    float thr  = thr_acc / 81.0f;
    float gate = -thr - 1.0f;

    int n_acc = BATCH * NSEG * NCLS;
    init_kernel<<<dim3((n_acc + 255) / 256), 256, 0, stream>>>(counts_g, probsum_g, cand_cnt, n_acc);
    vote_kernel<<<dim3(WW/32, HH/8, BATCH), dim3(32, 8), 0, stream>>>(regs, vote);
    pool_kernel<<<dim3(WW/32, HH/8, BATCH), dim3(32, 8), 0, stream>>>(
        vote, center_map, cand_vals, cand_idx, cand_cnt, gate);
    topk_kernel<<<dim3(BATCH), 256, 0, stream>>>(cand_vals, cand_idx, cand_cnt,
                                                 top_vals, centers, bm4);
    nearest_center_kernel<<<dim3(HW/256, BATCH), 256, 0, stream>>>(regs, bm4, ccidx);
    seg_stats_kernel<<<dim3(1024, BATCH), 256, 0, stream>>>(
        logits, ccidx, inst_map, seg_map, counts_g, probsum_g);
    finalize_kernel<<<dim3(BATCH), 256, 0, stream>>>(counts_g, probsum_g,
                                                     inst_class, inst_prob, inst_size);
}